// CMSWrite_63814624084479
// MI455X (gfx1250) — compile-verified
//
#include <hip/hip_runtime.h>
#include <hip/hip_bf16.h>
#include <math.h>

typedef __attribute__((ext_vector_type(2))) float v2f;
typedef __attribute__((ext_vector_type(4))) float v4f;
typedef __attribute__((ext_vector_type(8))) float v8f;

#define LVL 4
#define NN  32768
#define DS  1024
#define DE  1024
#define DC  512
#define DK  128
#define DV  512
#define DZ  128
#define THRG 0.1f
#define EPSLN 1e-5f

// workspace layout (float offsets)
#define WS_SCORES 0
#define WS_K      (LVL*NN)              // 131072
#define WS_V      (WS_K + LVL*DK)       // +512
#define WS_WGT    (WS_V + LVL*DV)       // +2048
#define WS_KEEP   (WS_WGT + LVL)
#define WS_MAX    (WS_KEEP + LVL)
#define WS_COEF   (WS_MAX + LVL)

// ---------------------------------------------------------------------------
// Phase 1: per-level projections: x -> h -> (spectral==h*wr) -> LN -> g, v, k
// 4 blocks (one per level) x 128 threads. Cost is negligible.
// ---------------------------------------------------------------------------
__global__ __launch_bounds__(128)
void cms_proj_kernel(const float* __restrict__ s_t, const float* __restrict__ e_t,
                     const float* __restrict__ ctx, const float* __restrict__ decay,
                     const float* __restrict__ W1_0, const float* __restrict__ b1_0,
                     const float* __restrict__ W1_r, const float* __restrict__ b1_r,
                     const float* __restrict__ spec_wr,
                     const float* __restrict__ ln_g, const float* __restrict__ ln_b,
                     const float* __restrict__ Wg, const float* __restrict__ bg,
                     const float* __restrict__ Wv, const float* __restrict__ bv,
                     const float* __restrict__ Wk, const float* __restrict__ bk,
                     float* __restrict__ ws)
{
    __shared__ float xs[DS + DC + DE];   // 2560 floats
    __shared__ float zs[DZ];
    __shared__ float red[DZ];
    const int l = blockIdx.x;
    const int t = threadIdx.x;           // 0..127
    const int xdim = (l == 0) ? (DS + DE) : (DS + DC + DE);

    // assemble x in LDS
    for (int j = t; j < DS; j += 128) xs[j] = s_t[j];
    if (l == 0) {
        for (int j = t; j < DE; j += 128) xs[DS + j] = e_t[j];
    } else {
        for (int j = t; j < DC; j += 128) xs[DS + j] = ctx[(l - 1) * DC + j];
        for (int j = t; j < DE; j += 128) xs[DS + DC + j] = e_t[j];
    }
    __syncthreads();

    // h[t] = W1 row t . x + b1[t]
    const float* Wrow;
    float acc;
    if (l == 0) {
        Wrow = W1_0 + (size_t)t * (DS + DE);
        acc  = b1_0[t];
    } else {
        Wrow = W1_r + (size_t)(l - 1) * DZ * (DS + DC + DE) + (size_t)t * (DS + DC + DE);
        acc  = b1_r[(l - 1) * DZ + t];
    }
    for (int j = 0; j < xdim; j += 4) {
        v4f w = *(const v4f*)(Wrow + j);
        acc += w.x * xs[j] + w.y * xs[j + 1] + w.z * xs[j + 2] + w.w * xs[j + 3];
    }
    // spectral block degenerates to elementwise * spec_wr (len-1 FFT identity)
    float y = acc * spec_wr[l * DZ + t];

    // LayerNorm over 128
    red[t] = y; __syncthreads();
    for (int s = 64; s > 0; s >>= 1) { if (t < s) red[t] += red[t + s]; __syncthreads(); }
    const float mu = red[0] * (1.0f / DZ);
    __syncthreads();
    const float d = y - mu;
    red[t] = d * d; __syncthreads();
    for (int s = 64; s > 0; s >>= 1) { if (t < s) red[t] += red[t + s]; __syncthreads(); }
    const float var = red[0] * (1.0f / DZ);
    const float z = d * rsqrtf(var + EPSLN) * ln_g[l * DZ + t] + ln_b[l * DZ + t];
    __syncthreads();
    zs[t]  = z;
    red[t] = Wg[l * DZ + t] * z;
    __syncthreads();
    for (int s = 64; s > 0; s >>= 1) { if (t < s) red[t] += red[t + s]; __syncthreads(); }
    if (t == 0) {
        const float g = 1.0f / (1.0f + __expf(-(red[0] + bg[l])));
        ws[WS_WGT  + l] = (g >= THRG) ? g : 0.0f;
        ws[WS_KEEP + l] = 1.0f - decay[l];
    }

    // k[t] = Wk row . z + bk
    {
        const float* wr = Wk + (size_t)(l * DZ + t) * DZ;
        float a = bk[l * DZ + t];
        for (int j = 0; j < DZ; j += 4) {
            v4f w = *(const v4f*)(wr + j);
            a += w.x * zs[j] + w.y * zs[j + 1] + w.z * zs[j + 2] + w.w * zs[j + 3];
        }
        ws[WS_K + l * DK + t] = a;
    }
    // v: 4 rows per thread
    for (int rr = 0; rr < 4; ++rr) {
        const int row = t + rr * 128;
        const float* wr = Wv + (size_t)(l * DV + row) * DZ;
        float a = bv[l * DV + row];
        for (int j = 0; j < DZ; j += 4) {
            v4f w = *(const v4f*)(wr + j);
            a += w.x * zs[j] + w.y * zs[j + 1] + w.z * zs[j + 2] + w.w * zs[j + 3];
        }
        ws[WS_V + l * DV + row] = tanhf(a);
    }
}

// ---------------------------------------------------------------------------
// Phase 2: scores = K_mem @ k / sqrt(DK) via V_WMMA_F32_16X16X4_F32.
// One wave -> 16 rows. A = 16x4 tile of K_mem (ISA A-layout: lanes 0-15 hold
// K=0,1; lanes 16-31 hold K=2,3). B = k chunk broadcast to all 16 columns, so
// every column of C carries the 16 scores. Lanes 0/16 extract M=0..7 / 8..15.
// ---------------------------------------------------------------------------
__global__ __launch_bounds__(256)
void cms_scores_kernel(const float* __restrict__ K_mem, float* __restrict__ ws)
{
    const int wave = (blockIdx.x * 256 + threadIdx.x) >> 5;  // tile id
    const int lane = threadIdx.x & 31;
    const int l    = wave >> 11;            // / (NN/16)
    const int row0 = (wave & 2047) << 4;
    const int half = lane >> 4;             // 0: K=0,1   1: K=2,3
    const int mrow = lane & 15;

    const float* Arow = K_mem + ((size_t)l * NN + row0 + mrow) * DK + half * 2;
    const float* kv   = ws + WS_K + l * DK + half * 2;

    v8f c = {};
#pragma unroll 4
    for (int j = 0; j < DK; j += 4) {
        v2f a = *(const v2f*)(Arow + j);
        v2f b = *(const v2f*)(kv + j);     // same value in all 16 lanes of the half
        c = __builtin_amdgcn_wmma_f32_16x16x4_f32(false, a, false, b,
                                                  (short)0, c, false, false);
    }
    const float scale = 0.08838834764831845f;  // 1/sqrt(128)
    if (mrow == 0) {  // lanes 0 and 16: hold M = half*8 + j in c[j]
        float* srow = ws + WS_SCORES + (size_t)l * NN + row0 + half * 8;
        v4f s0 = { c[0] * scale, c[1] * scale, c[2] * scale, c[3] * scale };
        v4f s1 = { c[4] * scale, c[5] * scale, c[6] * scale, c[7] * scale };
        *(v4f*)(srow)     = s0;
        *(v4f*)(srow + 4) = s1;
    }
}

// ---------------------------------------------------------------------------
// Phase 3: per-level softmax normalization (scores array = 128 KB/level).
// ---------------------------------------------------------------------------
__global__ __launch_bounds__(256)
void cms_softmax_kernel(float* __restrict__ ws)
{
    __shared__ float red[256];
    const int l = blockIdx.x;
    const int t = threadIdx.x;
    const float* sc = ws + WS_SCORES + (size_t)l * NN;

    float m = -3.402823466e38f;
    for (int i = t; i < NN; i += 256) m = fmaxf(m, sc[i]);
    red[t] = m; __syncthreads();
    for (int s = 128; s > 0; s >>= 1) { if (t < s) red[t] = fmaxf(red[t], red[t + s]); __syncthreads(); }
    m = red[0];
    __syncthreads();
    float sum = 0.0f;
    for (int i = t; i < NN; i += 256) sum += __expf(sc[i] - m);
    red[t] = sum; __syncthreads();
    for (int s = 128; s > 0; s >>= 1) { if (t < s) red[t] += red[t + s]; __syncthreads(); }
    if (t == 0) {
        ws[WS_MAX  + l] = m;
        ws[WS_COEF + l] = ws[WS_WGT + l] / red[0];   // wgt * (1/sum)
    }
}

// ---------------------------------------------------------------------------
// Phase 4: streaming update. One wave per row (L*N rows). NT load of M,
// NT store of out; K_mem read hits L2 (16MB/level, resident from phase 2).
// ---------------------------------------------------------------------------
__global__ __launch_bounds__(256)
void cms_update_kernel(const float* __restrict__ M, const float* __restrict__ K_mem,
                       const float* __restrict__ ws, float* __restrict__ out)
{
    const int  lane   = threadIdx.x & 31;
    const int  waveId = threadIdx.x >> 5;
    const long row    = (long)blockIdx.x * 8 + waveId;     // 0 .. L*N-1
    const int  l      = (int)(row >> 15);                  // / NN

    const float keep   = ws[WS_KEEP + l];
    const float mshift = ws[WS_MAX  + l];
    const float cbase  = ws[WS_COEF + l];
    const float score  = ws[WS_SCORES + row];
    const float coef   = cbase * __expf(score - mshift);   // wgt * alpha[i]

    const float* vsrc = ws + WS_V + l * DV;
    const float* ksrc = ws + WS_K + l * DK;
    const float* Mrow = M     + row * DV;
    const float* Krow = K_mem + row * DK;
    float*       orow = out   + row * (DV + DK);

#pragma unroll
    for (int c = 0; c < 4; ++c) {
        const int off = c * 128 + lane * 4;
        v4f mv = __builtin_nontemporal_load((const v4f*)(Mrow + off));
        v4f vv = *(const v4f*)(vsrc + off);
        v4f r  = keep * mv + coef * vv;
        __builtin_nontemporal_store(r, (v4f*)(orow + off));
    }
    {
        const int off = lane * 4;
        v4f kv = *(const v4f*)(Krow + off);
        v4f kk = *(const v4f*)(ksrc + off);
        v4f r  = keep * kv + coef * kk;
        __builtin_nontemporal_store(r, (v4f*)(orow + DV + off));
    }
}

extern "C" void kernel_launch(void* const* d_in, const int* in_sizes, int n_in,
                              void* d_out, int out_size, void* d_ws, size_t ws_size,
                              hipStream_t stream) {
    const float* s_t     = (const float*)d_in[0];
    const float* e_t     = (const float*)d_in[1];
    const float* ctx     = (const float*)d_in[2];
    const float* M       = (const float*)d_in[3];
    const float* K_mem   = (const float*)d_in[4];
    const float* decay   = (const float*)d_in[5];
    const float* W1_0    = (const float*)d_in[6];
    const float* b1_0    = (const float*)d_in[7];
    const float* W1_r    = (const float*)d_in[8];
    const float* b1_r    = (const float*)d_in[9];
    const float* spec_wr = (const float*)d_in[10];
    // d_in[11] = spec_wi: unused (len-1 irfft keeps only the real part)
    const float* ln_g    = (const float*)d_in[12];
    const float* ln_b    = (const float*)d_in[13];
    const float* Wg      = (const float*)d_in[14];
    const float* bg      = (const float*)d_in[15];
    const float* Wv      = (const float*)d_in[16];
    const float* bv      = (const float*)d_in[17];
    const float* Wk      = (const float*)d_in[18];
    const float* bk      = (const float*)d_in[19];
    float* out = (float*)d_out;
    float* ws  = (float*)d_ws;

    cms_proj_kernel<<<LVL, 128, 0, stream>>>(s_t, e_t, ctx, decay, W1_0, b1_0,
                                             W1_r, b1_r, spec_wr, ln_g, ln_b,
                                             Wg, bg, Wv, bv, Wk, bk, ws);
    cms_scores_kernel<<<(LVL * NN / 16) / 8, 256, 0, stream>>>(K_mem, ws);
    cms_softmax_kernel<<<LVL, 256, 0, stream>>>(ws);
    cms_update_kernel<<<(LVL * NN) / 8, 256, 0, stream>>>(M, K_mem, ws, out);
}